// FilteredDownsample2D_91053306675842
// MI455X (gfx1250) — compile-verified
//
#include <hip/hip_runtime.h>

typedef __attribute__((ext_vector_type(2))) float v2f;
typedef __attribute__((ext_vector_type(8))) float v8f;

#define TILE_IN    169   // input footprint of a 16x16 output tile (8*15+49)
#define ROWS_P     176   // rows padded to 11 bands of 16
#define COLS_P     172   // K padded to 43 chunks of 4 (172%64=44 -> conflict-free)
#define NKC        43    // K-chunks per band
#define MID_STRIDE 17    // mid LDS row stride (conflict-free)
#define PAD1       20

// smem floats: in + B(lane layout) + mid + k1
#define SMEM_FLOATS (ROWS_P * COLS_P + NKC * 64 + ROWS_P * MID_STRIDE + 64)

__device__ __forceinline__ int reflect512(int i) {
    i = (i < 0) ? -i : i;
    i = (i >= 512) ? (1022 - i) : i;
    return i;
}

__global__ void FilteredDownsample2D_kernel(const float* __restrict__ x,
                                            const float* __restrict__ filt,
                                            float* __restrict__ out) {
    extern __shared__ float smem[];
    float* sIn  = smem;                                   // ROWS_P * COLS_P
    float* sB   = sIn + ROWS_P * COLS_P;                  // NKC * 64 (kc,lane,2)
    float* sMid = sB + NKC * 64;                          // ROWS_P * MID_STRIDE
    float* sK1  = sMid + ROWS_P * MID_STRIDE;             // 64 (49 used)

    const int tid  = threadIdx.x;
    const int lane = tid & 31;
    const int wid  = tid >> 5;

    const int img = blockIdx.y;          // n*64 + c, 0..511
    const int oyb = blockIdx.x >> 2;     // 0..3
    const int oxb = blockIdx.x & 3;      // 0..3
    const int y0  = oyb * 128 - PAD1;    // tile origin in unpadded coords
    const int x0  = oxb * 128 - PAD1;

    // --- 1) recover 1D kernel: k1[j] = sum_i filt[i][j] (filt = k1 (x) k1, sum k1 = 1)
    if (tid < 49) {
        float s = 0.f;
        #pragma unroll 7
        for (int i = 0; i < 49; ++i) s += filt[i * 49 + tid];
        sK1[tid] = s;
    }
    __syncthreads();

    // --- 2a) banded B matrix in WMMA B-operand lane layout:
    //         B[t][j] = k1[t - 8j] for 0 <= t-8j < 49, else 0
    //         lanes 0-15 hold K-subrows {0,1}, lanes 16-31 hold {2,3}
    for (int e = tid; e < NKC * 64; e += 256) {
        int kc    = e >> 6;
        int r     = e & 63;
        int l     = r >> 1;
        int which = r & 1;
        int t = kc * 4 + ((l < 16) ? which : (2 + which));
        int j = l & 15;
        int d = t - 8 * j;
        sB[e] = (d >= 0 && d < 49) ? sK1[d] : 0.f;
    }

    // --- 2b) cooperative reflect-padded tile load, 8 rows x 32 lanes per pass.
    //         Each lane precomputes its 6 reflected x-coords once; one reflected
    //         y per row step. Steady state = global b32 load + ds store.
    {
        const float* xi = x + (size_t)img * (512 * 512);
        int   xoff[6];
        bool  cok[6];
        #pragma unroll
        for (int i = 0; i < 6; ++i) {
            int c = i * 32 + lane;
            cok[i]  = (c < TILE_IN);             // cols 169..171 are zero pad
            xoff[i] = reflect512(x0 + (cok[i] ? c : 0));
        }
        for (int rbase = 0; rbase < ROWS_P; rbase += 8) {
            int r = rbase + wid;
            bool rok = (r < TILE_IN);            // rows 169..175 are zero pad
            const float* rowp = xi + (size_t)reflect512(y0 + (rok ? r : 0)) * 512;
            float* dst = sIn + r * COLS_P;
            #pragma unroll
            for (int i = 0; i < 6; ++i) {
                int c = i * 32 + lane;
                if (c < COLS_P) {
                    float v = (rok && cok[i]) ? rowp[xoff[i]] : 0.f;
                    dst[c] = v;
                }
            }
        }
    }
    __syncthreads();

    // --- 3) horizontal pass via f32 WMMA. B is hoisted into 86 VGPRs (shared by
    //        every band this wave computes); fully unrolled K loop gives A loads
    //        constant ds offsets: 1 ds_load_b64 + 1 v_wmma per K-chunk.
    const int colA0 = (lane < 16) ? 0 : 2;
    const int rowA  = lane & 15;

    v2f bReg[NKC];
    {
        const float* bBase = sB + lane * 2;
        #pragma unroll
        for (int kc = 0; kc < NKC; ++kc)
            bReg[kc] = *(const v2f*)(bBase + kc * 64);
    }

    for (int band = wid; band < 11; band += 8) {
        v8f acc = {};
        const float* aBase = sIn + (band * 16 + rowA) * COLS_P + colA0;
        #pragma unroll
        for (int kc = 0; kc < NKC; ++kc) {
            v2f a = *(const v2f*)(aBase + kc * 4);
            acc = __builtin_amdgcn_wmma_f32_16x16x4_f32(
                false, a, false, bReg[kc], (short)0, acc, false, false);
        }
        // D layout: VGPR v of lane L -> row (v + (L<16?0:8)), col L%16
        int mbase = band * 16 + ((lane < 16) ? 0 : 8);
        int jcol  = lane & 15;
        #pragma unroll
        for (int v = 0; v < 8; ++v)
            sMid[(mbase + v) * MID_STRIDE + jcol] = acc[v];
    }
    __syncthreads();

    // --- 4) vertical 49-tap pass (VALU, ~11% of FLOPs) + coalesced store
    const int oyl = tid >> 4;
    const int oxl = tid & 15;
    float r0 = 0.f;
    int base = (8 * oyl) * MID_STRIDE + oxl;
    #pragma unroll 7
    for (int tp = 0; tp < 49; ++tp)
        r0 = fmaf(sMid[base + tp * MID_STRIDE], sK1[tp], r0);
    const int oy = oyb * 16 + oyl;
    const int ox = oxb * 16 + oxl;
    out[((size_t)img * 64 + oy) * 64 + ox] = r0;
}

extern "C" void kernel_launch(void* const* d_in, const int* in_sizes, int n_in,
                              void* d_out, int out_size, void* d_ws, size_t ws_size,
                              hipStream_t stream) {
    (void)in_sizes; (void)n_in; (void)d_ws; (void)ws_size; (void)out_size;
    const float* x    = (const float*)d_in[0];  // (8,64,512,512) fp32
    const float* filt = (const float*)d_in[1];  // (49,49) fp32
    float* out = (float*)d_out;                 // (8,64,64,64) fp32

    dim3 grid(16, 512);  // 4x4 output tiles per image, 512 channel-images
    size_t smem = SMEM_FLOATS * sizeof(float);  // 144,320 bytes
    FilteredDownsample2D_kernel<<<grid, 256, smem, stream>>>(x, filt, out);
}